// PIDSE_44530220925582
// MI455X (gfx1250) — compile-verified
//
#include <hip/hip_runtime.h>

#define DT_STEP 0.01f

typedef __attribute__((ext_vector_type(2))) float v2f;
typedef __attribute__((ext_vector_type(8))) float v8f;

// C[16 x N] = A[16 x K] @ B[K x N] using V_WMMA_F32_16X16X4_F32.
// A: LDS row-major (lda). B: either global row-major with zero-pad guards
// (GLOBB, actual dims Kact x Nact, ldb = Nact) or LDS (TRANSB reads B^T).
// C: LDS row-major (ldc). N-tiles are distributed across waves.
template<bool GLOBB, bool TRANSB>
__device__ __forceinline__ void wave_mm(
    const float* A, int lda,
    const float* B, int ldb, int Kact, int Nact,
    float* C, int ldc, int K, int N,
    int wave, int nwaves, int lane)
{
  const int row = lane & 15;     // A row / C col
  const int hi  = lane >> 4;     // half-wave selects K pair / C row block
  for (int n0 = wave * 16; n0 < N; n0 += nwaves * 16) {
    v8f acc = {};
    for (int k = 0; k < K; k += 4) {
      const int ka = k + 2 * hi;
      const int col = n0 + row;
      v2f a, bb;
      a.x = A[row * lda + ka];
      a.y = A[row * lda + ka + 1];
      if (GLOBB) {
        bb.x = (ka     < Kact && col < Nact) ? B[(ka    ) * ldb + col] : 0.0f;
        bb.y = (ka + 1 < Kact && col < Nact) ? B[(ka + 1) * ldb + col] : 0.0f;
      } else if (TRANSB) {
        bb.x = B[col * ldb + ka];
        bb.y = B[col * ldb + ka + 1];
      } else {
        bb.x = B[(ka    ) * ldb + col];
        bb.y = B[(ka + 1) * ldb + col];
      }
      acc = __builtin_amdgcn_wmma_f32_16x16x4_f32(
          /*neg_a=*/false, a, /*neg_b=*/false, bb,
          /*c_mod=*/(short)0, acc, /*reuse_a=*/false, /*reuse_b=*/false);
    }
#pragma unroll
    for (int v = 0; v < 8; ++v)
      C[(hi * 8 + v) * ldc + n0 + row] = acc[v];
  }
}

// Row 0 = value row: a = act(pre + bias). Rows 1..12 = Jacobian tangents:
// scaled by act'(pre). Rows 13..15 already zero (A rows were zero).
__device__ __forceinline__ void aug_act(float* M, int ld, const float* bias,
                                        int N, int tid, int nth, bool use_tanh)
{
  for (int c = tid; c < N; c += nth) {
    float p = M[c] + bias[c];
    float a, g;
    if (use_tanh) { a = tanhf(p); g = 1.0f - a * a; }
    else          { a = p > 0.0f ? p : 0.0f; g = p > 0.0f ? 1.0f : 0.0f; }
    M[c] = a;
#pragma unroll
    for (int r = 1; r <= 12; ++r) M[r * ld + c] *= g;
  }
}

__global__ __launch_bounds__(128) void ekf_wmma_kernel(
    const float* __restrict__ ctrl, const float* __restrict__ meas,
    const float* __restrict__ x0g, const float* __restrict__ P0g,
    const float* __restrict__ Qg, const float* __restrict__ Rg,
    const float* __restrict__ dW0, const float* __restrict__ db0,
    const float* __restrict__ dW1, const float* __restrict__ db1,
    const float* __restrict__ dW2, const float* __restrict__ db2,
    const float* __restrict__ dW3, const float* __restrict__ db3,
    const float* __restrict__ mW0, const float* __restrict__ mb0,
    const float* __restrict__ mW1, const float* __restrict__ mb1,
    const float* __restrict__ mW2, const float* __restrict__ mb2,
    float* __restrict__ outE, float* __restrict__ outZ, float* __restrict__ outP,
    int T)
{
  __shared__ float smem[3520];       // 14.1 KB: low LDS -> high occupancy
  float* XA   = smem;                // 16x65 activation ping
  float* XB   = smem + 1040;         // 16x65 activation pong
  float* P    = smem + 2080;         // 16x17 covariance (padded 12x12)
  float* Pp   = smem + 2352;         // 16x17 predicted covariance
  float* F    = smem + 2624;         // 16x17
  float* H    = smem + 2896;         // 16x17
  float* Km   = smem + 3168;         // 16x17 Kalman gain
  float* xcur = smem + 3440;
  float* xp   = smem + 3456;
  float* zp   = smem + 3472;
  float* yv   = smem + 3488;
  float* xnew = smem + 3504;
  // pool aliases (lifetimes verified against XA/XB phases)
  float* T1   = smem + 1040;         // F@P scratch (XB region, free then)
  float* MOUT = smem + 1040;         // measurement final layer output
  float* HP   = smem + 0;            // H@P
  float* Sm   = smem + 288;          // innovation covariance
  float* PHt  = smem + 576;          // P@H^T
  float* AUG  = smem + 864;          // 9x24 Gauss-Jordan [S | PHt^T]
  float* rowp = smem + 1088;
  float* fac  = smem + 1120;
  float* KH   = smem + 1152;
  float* Pn   = smem + 1440;

  const int b    = blockIdx.x;
  const int tid  = threadIdx.x;
  const int lane = tid & 31;
  const int wave = tid >> 5;
  const int NTH = 128, NW = 4;

  if (tid < 12) xcur[tid] = x0g[b * 12 + tid];
  for (int idx = tid; idx < 256; idx += NTH) {
    int r = idx >> 4, c = idx & 15;
    P[r * 17 + c] = (r < 12 && c < 12) ? P0g[b * 144 + r * 12 + c] : 0.0f;
  }
  __syncthreads();

  for (int t = 0; t < T; ++t) {
    const float* x_in;
    const float* P_in;
    if (t > 0) {
      // ================= PREDICT =================
      // Stage augmented [ [x,u] ; I12 ; 0 ] (16x16)
      for (int idx = tid; idx < 256; idx += NTH) {
        int r = idx >> 4, c = idx & 15;
        float v;
        if (r == 0)       v = (c < 12) ? xcur[c]
                                       : ctrl[((size_t)b * T + t) * 4 + (c - 12)];
        else if (r <= 12) v = (c == r - 1) ? 1.0f : 0.0f;
        else              v = 0.0f;
        XA[r * 65 + c] = v;
      }
      __syncthreads();
      wave_mm<true,false>(XA,65, dW0,64, 16,64, XB,65, 16,64, wave,NW,lane);
      __syncthreads();
      aug_act(XB,65, db0, 64, tid,NTH, true);
      __syncthreads();
      wave_mm<true,false>(XB,65, dW1,64, 64,64, XA,65, 64,64, wave,NW,lane);
      __syncthreads();
      aug_act(XA,65, db1, 64, tid,NTH, true);
      __syncthreads();
      wave_mm<true,false>(XA,65, dW2,32, 64,32, XB,65, 64,32, wave,NW,lane);
      __syncthreads();
      aug_act(XB,65, db2, 32, tid,NTH, true);
      __syncthreads();
      wave_mm<true,false>(XB,65, dW3,12, 32,12, XA,65, 32,16, wave,NW,lane);
      __syncthreads();
      if (tid < 12) XA[tid] += db3[tid];         // bias on value row only
      __syncthreads();
      // F = I + DT*D + J  (J[i][j] = tangent row j, col i); x_pred
      for (int idx = tid; idx < 256; idx += NTH) {
        int i = idx >> 4, j = idx & 15;
        float v = 0.0f;
        if (i < 12 && j < 12) {
          v = XA[(j + 1) * 65 + i];
          if (i == j) v += 1.0f;
          if (((i < 3) || (i >= 6 && i < 9)) && (j == i + 3)) v += DT_STEP;
        }
        F[i * 17 + j] = v;
      }
      if (tid < 12) {
        float drift = (tid < 3 || (tid >= 6 && tid < 9)) ? xcur[tid + 3] : 0.0f;
        xp[tid] = xcur[tid] + DT_STEP * drift + XA[tid];
      }
      __syncthreads();
      // P_pred = F P F^T + Q
      wave_mm<false,false>(F,17, P,17, 16,16, T1,17, 16,16, wave,NW,lane);
      __syncthreads();
      wave_mm<false,true >(T1,17, F,17, 16,16, Pp,17, 16,16, wave,NW,lane);
      __syncthreads();
      for (int idx = tid; idx < 144; idx += NTH)
        Pp[(idx / 12) * 17 + (idx % 12)] += Qg[idx];
      __syncthreads();
      x_in = xp; P_in = Pp;
    } else {
      x_in = xcur; P_in = P;
    }

    // ================= UPDATE =================
    for (int idx = tid; idx < 256; idx += NTH) {
      int r = idx >> 4, c = idx & 15;
      float v;
      if (r == 0)       v = (c < 12) ? x_in[c] : 0.0f;
      else if (r <= 12) v = (c == r - 1) ? 1.0f : 0.0f;
      else              v = 0.0f;
      XA[r * 65 + c] = v;
    }
    __syncthreads();
    wave_mm<true,false>(XA,65, mW0,64, 12,64, XB,65, 16,64, wave,NW,lane);
    __syncthreads();
    aug_act(XB,65, mb0, 64, tid,NTH, false);
    __syncthreads();
    wave_mm<true,false>(XB,65, mW1,64, 64,64, XA,65, 64,64, wave,NW,lane);
    __syncthreads();
    aug_act(XA,65, mb1, 64, tid,NTH, false);
    __syncthreads();
    wave_mm<true,false>(XA,65, mW2,9, 64,9, MOUT,17, 64,16, wave,NW,lane);
    __syncthreads();
    if (tid < 9) zp[tid] = x_in[tid] + MOUT[tid] + mb2[tid];
    for (int idx = tid; idx < 256; idx += NTH) {
      int i = idx >> 4, j = idx & 15;
      float v = 0.0f;
      if (i < 9 && j < 12) {
        v = MOUT[(j + 1) * 17 + i];
        if (i == j) v += 1.0f;
      }
      H[i * 17 + j] = v;
    }
    __syncthreads();
    wave_mm<false,false>(H,17, P_in,17, 16,16, HP,17, 16,16, wave,NW,lane);
    __syncthreads();
    if (wave == 0)      wave_mm<false,true>(HP,17,  H,17, 16,16, Sm,17, 16,16, 0,1,lane);
    else if (wave == 1) wave_mm<false,true>(P_in,17,H,17, 16,16, PHt,17,16,16, 0,1,lane);
    __syncthreads();
    for (int idx = tid; idx < 81; idx += NTH)
      Sm[(idx / 9) * 17 + (idx % 9)] += Rg[idx];
    __syncthreads();
    // AUG = [S | PHt^T]  (9 x 21)
    for (int idx = tid; idx < 9 * 21; idx += NTH) {
      int r = idx / 21, c = idx % 21;
      AUG[r * 24 + c] = (c < 9) ? Sm[r * 17 + c] : PHt[(c - 9) * 17 + r];
    }
    __syncthreads();
    // Gauss-Jordan (S is SPD: no pivoting)
    for (int p = 0; p < 9; ++p) {
      if (tid < 21) rowp[tid] = AUG[p * 24 + tid] / AUG[p * 24 + p];
      __syncthreads();
      if (tid < 9) fac[tid] = AUG[tid * 24 + p];
      __syncthreads();
      for (int idx = tid; idx < 9 * 21; idx += NTH) {
        int r = idx / 21, c = idx % 21;
        AUG[r * 24 + c] = (r == p) ? rowp[c] : (AUG[r * 24 + c] - fac[r] * rowp[c]);
      }
      __syncthreads();
    }
    // K[i][j] = AUG[j][9+i]
    for (int idx = tid; idx < 256; idx += NTH) {
      int i = idx >> 4, j = idx & 15;
      Km[i * 17 + j] = (i < 12 && j < 9) ? AUG[j * 24 + 9 + i] : 0.0f;
    }
    if (tid < 9) yv[tid] = meas[((size_t)b * T + t) * 9 + tid] - zp[tid];
    __syncthreads();
    if (tid < 12) {
      float s = x_in[tid];
#pragma unroll
      for (int j = 0; j < 9; ++j) s += Km[tid * 17 + j] * yv[j];
      xnew[tid] = s;
    }
    wave_mm<false,false>(Km,17, H,17, 16,16, KH,17, 16,16, wave,NW,lane);
    __syncthreads();
    for (int idx = tid; idx < 256; idx += NTH) {
      int i = idx >> 4, j = idx & 15;
      float v = 0.0f;
      if (i < 12 && j < 12) v = ((i == j) ? 1.0f : 0.0f) - KH[i * 17 + j];
      KH[i * 17 + j] = v;
    }
    __syncthreads();
    wave_mm<false,false>(KH,17, P_in,17, 16,16, Pn,17, 16,16, wave,NW,lane);
    __syncthreads();
    // outputs + commit state
    size_t bt = (size_t)b * T + t;
    if (tid < 12) outE[bt * 12 + tid] = xnew[tid];
    if (tid < 9)  outZ[bt * 9 + tid]  = zp[tid];
    for (int idx = tid; idx < 144; idx += NTH)
      outP[bt * 144 + idx] = Pn[(idx / 12) * 17 + (idx % 12)];
    if (tid < 12) xcur[tid] = xnew[tid];
    for (int idx = tid; idx < 256; idx += NTH)
      P[idx / 16 * 17 + (idx & 15)] = Pn[idx / 16 * 17 + (idx & 15)];
    if (tid == 0 && t + 1 < T) {
      __builtin_prefetch(&meas[((size_t)b * T + t + 1) * 9], 0, 0);
      __builtin_prefetch(&ctrl[((size_t)b * T + t + 1) * 4], 0, 0);
    }
    __syncthreads();
  }
}

extern "C" void kernel_launch(void* const* d_in, const int* in_sizes, int n_in,
                              void* d_out, int out_size, void* d_ws, size_t ws_size,
                              hipStream_t stream) {
  (void)n_in; (void)out_size; (void)d_ws; (void)ws_size;
  // inputs: 0=states(unused),1=controls,2=measurements,3=initial_state,
  // 4=initial_covariance,5=Q,6=R,7..14=dW0,db0..dW3,db3,15..20=mW0,mb0..mW2,mb2
  const float* ctrl = (const float*)d_in[1];
  const float* meas = (const float*)d_in[2];
  const float* x0   = (const float*)d_in[3];
  const float* P0   = (const float*)d_in[4];
  const float* Q    = (const float*)d_in[5];
  const float* R    = (const float*)d_in[6];
  const int B = in_sizes[3] / 12;
  const int T = in_sizes[1] / (B * 4);
  float* outE = (float*)d_out;
  float* outZ = outE + (size_t)B * T * 12;
  float* outP = outZ + (size_t)B * T * 9;
  ekf_wmma_kernel<<<B, 128, 0, stream>>>(
      ctrl, meas, x0, P0, Q, R,
      (const float*)d_in[7],  (const float*)d_in[8],
      (const float*)d_in[9],  (const float*)d_in[10],
      (const float*)d_in[11], (const float*)d_in[12],
      (const float*)d_in[13], (const float*)d_in[14],
      (const float*)d_in[15], (const float*)d_in[16],
      (const float*)d_in[17], (const float*)d_in[18],
      (const float*)d_in[19], (const float*)d_in[20],
      outE, outZ, outP, T);
}